// GraphNorm_6811818131525
// MI455X (gfx1250) — compile-verified
//
#include <hip/hip_runtime.h>

typedef __attribute__((ext_vector_type(2))) float v2f;
typedef __attribute__((ext_vector_type(4))) float v4f;
typedef __attribute__((ext_vector_type(8))) float v8f;

#define GN_EPS 1e-5f
#define G_MAX 512   // >= actual batch_size (256); indices masked for safety

// ---------------------------------------------------------------------------
// Pass 1: per-segment sums of x and x*x, plus per-segment counts.
// One block = rowsPerBlock sorted rows; wave w handles features [16w,16w+16).
// Bulk path: 16 rows per iteration = 4x V_WMMA_F32_16X16X4_F32 pairs with
// A = ones (column sums accumulate in C). Segment boundaries handled by a
// 4-row path and a scalar path; partial sums flushed with f32 atomics.
// FC > 0 specializes the feature stride at compile time so the 8 loads of the
// fast path use one address register + immediate offsets.
// ---------------------------------------------------------------------------
template <int FC>
__global__ __launch_bounds__(256) void gn_stats(const float* __restrict__ x,
                                                const int* __restrict__ batch,
                                                float* __restrict__ s1,
                                                float* __restrict__ s2,
                                                int* __restrict__ counts,
                                                int N, int Fdyn, int rowsPerBlock)
{
    const int F = (FC > 0) ? FC : Fdyn;

    __shared__ int hist[G_MAX];
    const int tid = threadIdx.x;
    const int nth = blockDim.x;
    const int r0  = blockIdx.x * rowsPerBlock;
    const int r1  = min(r0 + rowsPerBlock, N);

    // ---- counts via LDS histogram (sorted batch -> ~1-2 hot bins/block) ----
    for (int g = tid; g < G_MAX; g += nth) hist[g] = 0;
    __syncthreads();
    for (int r = r0 + tid; r < r1; r += nth) {
        int s = batch[r] & (G_MAX - 1);
        atomicAdd(&hist[s], 1);
    }
    __syncthreads();
    for (int g = tid; g < G_MAX; g += nth) {
        int h = hist[g];
        if (h) atomicAdd(&counts[g], h);
    }

    if (r0 >= r1) return;

    // ---- per-wave WMMA segment reduction ----
    const int wave  = tid >> 5;
    const int lane  = tid & 31;
    const int col   = wave * 16 + (lane & 15);
    const int khalf = (lane >> 4) << 1;   // B layout: VGPR0 = K0 (lanes 0-15) / K2 (lanes 16-31)

    const v8f zero = {0.f, 0.f, 0.f, 0.f, 0.f, 0.f, 0.f, 0.f};
    v8f c1 = zero, c2 = zero;             // C accumulators: row 0 = running feature sums
    v2f ones; ones.x = 1.f; ones.y = 1.f; // A = ones(16x4): D[m,n] = sum_k B[k,n] + C[m,n]

    int r   = r0;
    int cur = batch[r0] & (G_MAX - 1);    // wave-uniform (uniform-address load)
    const float* p = x + (size_t)(r0 + khalf) * F + col;  // running base pointer

    while (r < r1) {
        // wave-uniform branches: EXEC stays all-ones around WMMA as required
        if ((r + 16 <= r1) && ((batch[r + 15] & (G_MAX - 1)) == cur)) {
            // 16 rows, same segment: 8 loads (immediate offsets) + 4 WMMA pairs
#pragma unroll
            for (int b = 0; b < 4; ++b) {
                v2f bv;
                bv.x = p[(size_t)(4 * b) * F];
                bv.y = p[(size_t)(4 * b + 1) * F];
                v2f bq = bv * bv;
                c1 = __builtin_amdgcn_wmma_f32_16x16x4_f32(false, ones, false, bv,
                                                           (short)0, c1, false, false);
                c2 = __builtin_amdgcn_wmma_f32_16x16x4_f32(false, ones, false, bq,
                                                           (short)0, c2, false, false);
            }
            r += 16;
            p += (size_t)16 * F;
        } else if ((r + 4 <= r1) && ((batch[r + 3] & (G_MAX - 1)) == cur)) {
            v2f bv; bv.x = p[0]; bv.y = p[F];
            v2f bq = bv * bv;
            c1 = __builtin_amdgcn_wmma_f32_16x16x4_f32(false, ones, false, bv,
                                                       (short)0, c1, false, false);
            c2 = __builtin_amdgcn_wmma_f32_16x16x4_f32(false, ones, false, bq,
                                                       (short)0, c2, false, false);
            r += 4;
            p += (size_t)4 * F;
        } else {
            int s = batch[r] & (G_MAX - 1);
            if (s != cur) {
                if (lane < 16) {  // D[0,n] lives in C VGPR0, lanes 0-15
                    atomicAdd(&s1[(size_t)cur * F + col], c1[0]);
                    atomicAdd(&s2[(size_t)cur * F + col], c2[0]);
                }
                c1 = zero; c2 = zero;
                cur = s;
            } else {
                float v = x[(size_t)r * F + col];
                c1[0] += v;                    // only VGPR0 row matters for extraction
                c2[0]  = fmaf(v, v, c2[0]);
                ++r;
                p += F;
            }
        }
    }
    if (lane < 16) {
        atomicAdd(&s1[(size_t)cur * F + col], c1[0]);
        atomicAdd(&s2[(size_t)cur * F + col], c2[0]);
    }
}

// ---------------------------------------------------------------------------
// Pass 2: fold stats + gamma/beta into per-(segment,feature) scale/bias:
//   a = gamma * rsqrt(max(var,eps)+eps),  b = beta - mean*a
// ---------------------------------------------------------------------------
__global__ __launch_bounds__(256) void gn_finalize(const float* __restrict__ s1,
                                                   const float* __restrict__ s2,
                                                   const int* __restrict__ counts,
                                                   const float* __restrict__ gamma,
                                                   const float* __restrict__ beta,
                                                   float* __restrict__ Ac,
                                                   float* __restrict__ Bc,
                                                   int F)
{
    int idx = blockIdx.x * blockDim.x + threadIdx.x;
    if (idx >= G_MAX * F) return;
    int g = idx / F;
    int f = idx - g * F;
    float cnt  = fmaxf((float)counts[g], 1.0f);
    float inv  = 1.0f / cnt;
    float mean = s1[idx] * inv;
    float var  = fmaf(-mean, mean, s2[idx] * inv);
    var = fmaxf(var, GN_EPS);
    float rstd = rsqrtf(var + GN_EPS);
    float a = gamma[f] * rstd;
    Ac[idx] = a;
    Bc[idx] = fmaf(-mean, a, beta[f]);
}

// ---------------------------------------------------------------------------
// Pass 3: out = fma(x, a[seg], b[seg]).  float4 streaming, non-temporal on the
// 1 GB x read (last use) and 1 GB out write; a/b (256 KB) stay cache-resident.
// With Fq = F/4 = 32, batch[row] is wave-uniform -> scalar broadcast.
// ---------------------------------------------------------------------------
__global__ __launch_bounds__(256) void gn_norm(const v4f* __restrict__ x4,
                                               const int* __restrict__ batch,
                                               const v4f* __restrict__ A4,
                                               const v4f* __restrict__ B4,
                                               v4f* __restrict__ out4,
                                               long long totalQuads,
                                               int fqShift, int fqMask)
{
    long long idx = (long long)blockIdx.x * blockDim.x + threadIdx.x;
    if (idx >= totalQuads) return;
    long long row = idx >> fqShift;
    int fq  = (int)(idx & fqMask);
    int seg = batch[row] & (G_MAX - 1);
    size_t sb = ((size_t)seg << fqShift) + fq;

    v4f xv = __builtin_nontemporal_load(&x4[idx]);
    v4f av = A4[sb];
    v4f bv = B4[sb];
    v4f o;
    o.x = fmaf(xv.x, av.x, bv.x);
    o.y = fmaf(xv.y, av.y, bv.y);
    o.z = fmaf(xv.z, av.z, bv.z);
    o.w = fmaf(xv.w, av.w, bv.w);
    __builtin_nontemporal_store(o, &out4[idx]);
}

// ---------------------------------------------------------------------------
extern "C" void kernel_launch(void* const* d_in, const int* in_sizes, int n_in,
                              void* d_out, int out_size, void* d_ws, size_t ws_size,
                              hipStream_t stream)
{
    const float* x     = (const float*)d_in[0];
    const float* gamma = (const float*)d_in[1];
    const float* beta  = (const float*)d_in[2];
    const int*   batch = (const int*)d_in[3];
    // d_in[4] (batch_size) stays on device; buffers sized to G_MAX >= G.

    const int F = in_sizes[1];            // 128
    const int N = in_sizes[0] / F;        // 2,000,000

    // workspace layout: [s1][s2][counts][Ac][Bc]
    float* s1     = (float*)d_ws;
    float* s2     = s1 + (size_t)G_MAX * F;
    int*   counts = (int*)(s2 + (size_t)G_MAX * F);
    float* Ac     = (float*)(counts + G_MAX);
    float* Bc     = Ac + (size_t)G_MAX * F;

    size_t zeroBytes = (size_t)2 * G_MAX * F * sizeof(float) + G_MAX * sizeof(int);
    hipMemsetAsync(d_ws, 0, zeroBytes, stream);

    // Pass 1 (specialize on the common F=128 for immediate-offset codegen)
    const int rowsPerBlock = 512;
    int blocksA  = (N + rowsPerBlock - 1) / rowsPerBlock;
    int threadsA = 32 * (F / 16);         // one wave per 16 features (256 for F=128)
    if (F == 128) {
        gn_stats<128><<<blocksA, threadsA, 0, stream>>>(x, batch, s1, s2, counts,
                                                        N, F, rowsPerBlock);
    } else {
        gn_stats<0><<<blocksA, threadsA, 0, stream>>>(x, batch, s1, s2, counts,
                                                      N, F, rowsPerBlock);
    }

    // Pass 2
    int totB = G_MAX * F;
    gn_finalize<<<(totB + 255) / 256, 256, 0, stream>>>(s1, s2, counts,
                                                        gamma, beta, Ac, Bc, F);

    // Pass 3
    int Fq = F >> 2;
    int shift = 0; while ((1 << shift) < Fq) ++shift;   // F=128 -> Fq=32 -> shift=5
    long long totalQuads = (long long)N * Fq;
    long long blocksC = (totalQuads + 255) / 256;
    gn_norm<<<(unsigned)blocksC, 256, 0, stream>>>((const v4f*)x, batch,
                                                   (const v4f*)Ac, (const v4f*)Bc,
                                                   (v4f*)d_out,
                                                   totalQuads, shift, Fq - 1);
}